// RandDGMC_86483461472645
// MI455X (gfx1250) — compile-verified
//
#include <hip/hip_runtime.h>
#include <hip/hip_bf16.h>
#include <stdint.h>

// Problem constants (match reference)
#define N0V 10000
#define N1V 10000
#define KP1 11          // k+1 candidates
#define NSTEPS 10
#define RDIM 128
#define NEDGES 320000

typedef float v2f __attribute__((ext_vector_type(2)));
typedef float v8f __attribute__((ext_vector_type(8)));

// ---------------------------------------------------------------------------
// Threefry-2x32-20 (matches JAX rotation/key-injection schedule)
// ---------------------------------------------------------------------------
__device__ __forceinline__ void tf_round(uint32_t& x0, uint32_t& x1, int r) {
  x0 += x1;
  x1 = (x1 << r) | (x1 >> (32 - r));
  x1 ^= x0;
}

__device__ __forceinline__ void threefry2x32(uint32_t k0, uint32_t k1,
                                             uint32_t c0, uint32_t c1,
                                             uint32_t& o0, uint32_t& o1) {
  uint32_t ks2 = k0 ^ k1 ^ 0x1BD11BDAu;
  uint32_t x0 = c0 + k0, x1 = c1 + k1;
  tf_round(x0, x1, 13); tf_round(x0, x1, 15); tf_round(x0, x1, 26); tf_round(x0, x1, 6);
  x0 += k1; x1 += ks2 + 1u;
  tf_round(x0, x1, 17); tf_round(x0, x1, 29); tf_round(x0, x1, 16); tf_round(x0, x1, 24);
  x0 += ks2; x1 += k0 + 2u;
  tf_round(x0, x1, 13); tf_round(x0, x1, 15); tf_round(x0, x1, 26); tf_round(x0, x1, 6);
  x0 += k0; x1 += k1 + 3u;
  tf_round(x0, x1, 17); tf_round(x0, x1, 29); tf_round(x0, x1, 16); tf_round(x0, x1, 24);
  x0 += k1; x1 += ks2 + 4u;
  tf_round(x0, x1, 13); tf_round(x0, x1, 15); tf_round(x0, x1, 26); tf_round(x0, x1, 6);
  x0 += ks2; x1 += k0 + 5u;
  o0 = x0; o1 = x1;
}

// erfinv (Giles 2010 single precision approximation)
__device__ __forceinline__ float erfinv_f(float x) {
  float w = -__logf((1.0f - x) * (1.0f + x));
  float p;
  if (w < 5.0f) {
    w -= 2.5f;
    p = 2.81022636e-08f;
    p = fmaf(p, w, 3.43273939e-07f);
    p = fmaf(p, w, -3.5233877e-06f);
    p = fmaf(p, w, -4.39150654e-06f);
    p = fmaf(p, w, 0.00021858087f);
    p = fmaf(p, w, -0.00125372503f);
    p = fmaf(p, w, -0.00417768164f);
    p = fmaf(p, w, 0.246640727f);
    p = fmaf(p, w, 1.50140941f);
  } else {
    w = __fsqrt_rn(w) - 3.0f;
    p = -0.000200214257f;
    p = fmaf(p, w, 0.000100950558f);
    p = fmaf(p, w, 0.00134934322f);
    p = fmaf(p, w, -0.00367342844f);
    p = fmaf(p, w, 0.00573950773f);
    p = fmaf(p, w, -0.0076224613f);
    p = fmaf(p, w, 0.00943887047f);
    p = fmaf(p, w, 1.00167406f);
    p = fmaf(p, w, 2.83297682f);
  }
  return p * x;
}

__device__ __forceinline__ float normal_from_bits(uint32_t bits) {
  // JAX: floats in [1,2) via mantissa bits, -> [0,1), then affine to [lo,1), sqrt(2)*erfinv
  float f = __uint_as_float((bits >> 9) | 0x3f800000u) - 1.0f;
  const float lo = -0.99999994f, hi = 1.0f;
  float u = fmaf(f, hi - lo, lo);
  u = fmaxf(lo, u);
  return 1.41421356237f * erfinv_f(u);
}

// ---------------------------------------------------------------------------
// Kernel 1: row-wise top-11 of S_hat, write s_hat=vals*NSTEPS, knn idx, S=softmax
// One block (256 threads) per row; single streaming pass + LDS tree merge.
// ---------------------------------------------------------------------------
__global__ __launch_bounds__(256) void topk_kernel(const float* __restrict__ S_hat,
                                                   float* __restrict__ s_hat,
                                                   int* __restrict__ knn,
                                                   float* __restrict__ S) {
  __shared__ float lvA[256 * KP1], lvB[256 * KP1];
  __shared__ int   liA[256 * KP1], liB[256 * KP1];

  const int row = blockIdx.x;
  const int t = threadIdx.x;
  const float* rp = S_hat + (size_t)row * N1V;

  float lv[KP1];
  int   li[KP1];
#pragma unroll
  for (int q = 0; q < KP1; ++q) { lv[q] = -3.4e38f; li[q] = 0x7fffffff; }

  for (int c = t; c < N1V; c += 256) {
    if (c + 4096 < N1V) __builtin_prefetch(rp + c + 4096, 0, 1);
    float v = rp[c];
    if (v > lv[KP1 - 1]) {
      float cv = v; int ci = c;
#pragma unroll
      for (int q = 0; q < KP1; ++q) {
        if (cv > lv[q]) {
          float tv = lv[q]; int ti = li[q];
          lv[q] = cv; li[q] = ci;
          cv = tv; ci = ti;
        }
      }
    }
  }

#pragma unroll
  for (int q = 0; q < KP1; ++q) { lvA[t * KP1 + q] = lv[q]; liA[t * KP1 + q] = li[q]; }
  __syncthreads();

  float* sv = lvA; int* si = liA;
  float* dv = lvB; int* di = liB;
  for (int off = 128; off >= 1; off >>= 1) {
    if (t < off) {
      int a0 = t * KP1, b0 = (t + off) * KP1;
      int ia = 0, ib = 0;
#pragma unroll
      for (int q = 0; q < KP1; ++q) {
        float va = sv[a0 + ia], vb = sv[b0 + ib];
        int   xa = si[a0 + ia], xb = si[b0 + ib];
        bool takeA = (va > vb) || (va == vb && xa <= xb);
        dv[a0 + q] = takeA ? va : vb;
        di[a0 + q] = takeA ? xa : xb;
        if (takeA) ia++; else ib++;
      }
    }
    __syncthreads();
    float* tv = sv; sv = dv; dv = tv;
    int* ti = si; si = di; di = ti;
  }

  if (t == 0) {
    float sh[KP1];
    float mx = -3.4e38f;
#pragma unroll
    for (int q = 0; q < KP1; ++q) {
      sh[q] = sv[q] * (float)NSTEPS;
      mx = fmaxf(mx, sh[q]);
      knn[row * KP1 + q] = si[q];
      s_hat[row * KP1 + q] = sh[q];
    }
    float sum = 0.0f, ex[KP1];
#pragma unroll
    for (int q = 0; q < KP1; ++q) { ex[q] = __expf(sh[q] - mx); sum += ex[q]; }
    float inv = 1.0f / sum;
#pragma unroll
    for (int q = 0; q < KP1; ++q) S[row * KP1 + q] = ex[q] * inv;
  }
}

// ---------------------------------------------------------------------------
// Kernel 2: r_s = jax.random.normal(fold_in(key(42), step), (N0, 128))
// ---------------------------------------------------------------------------
__global__ __launch_bounds__(256) void rng_kernel(float* __restrict__ r_s, int step) {
  const int half = (N0V * RDIM) / 2;  // 640000
  int i = blockIdx.x * 256 + threadIdx.x;
  // fold_in(key(42), step) = threefry(key=(0,42), ctr=(0,step))
  uint32_t fk0, fk1;
  threefry2x32(0u, 42u, 0u, (uint32_t)step, fk0, fk1);
  uint32_t o0, o1;
  threefry2x32(fk0, fk1, (uint32_t)i, (uint32_t)(i + half), o0, o1);
  r_s[i] = normal_from_bits(o0);
  r_s[i + half] = normal_from_bits(o1);
}

// ---------------------------------------------------------------------------
// Kernel 3: r_t[knn[i,k]] += S[i,k] * r_s[i]   (one wave per (i,k) pair)
// ---------------------------------------------------------------------------
__global__ __launch_bounds__(256) void scatter_rt_kernel(const float* __restrict__ S,
                                                         const int* __restrict__ knn,
                                                         const float* __restrict__ r_s,
                                                         float* __restrict__ r_t) {
  int gid = blockIdx.x * 256 + threadIdx.x;
  int pair = gid >> 5;          // 0 .. N0*KP1-1
  int lane = gid & 31;
  int i = pair / KP1;
  int j = knn[pair];
  float s = S[pair];
  float4 v = ((const float4*)(r_s + (size_t)i * RDIM))[lane];
  float* dst = r_t + (size_t)j * RDIM + lane * 4;
  atomicAdd(dst + 0, s * v.x);
  atomicAdd(dst + 1, s * v.y);
  atomicAdd(dst + 2, s * v.z);
  atomicAdd(dst + 3, s * v.w);
}

// ---------------------------------------------------------------------------
// Kernel 4: agg[dst] += x[src] over edge list (one wave per edge)
// edges layout: [0:NEDGES] = src row, [NEDGES:2*NEDGES] = dst row
// ---------------------------------------------------------------------------
__global__ __launch_bounds__(256) void edge_scatter_kernel(const int* __restrict__ edges,
                                                           const float* __restrict__ x,
                                                           float* __restrict__ agg) {
  int gid = blockIdx.x * 256 + threadIdx.x;
  int e = gid >> 5;
  int lane = gid & 31;
  int src = edges[e];
  int dst = edges[NEDGES + e];
  float4 v = ((const float4*)(x + (size_t)src * RDIM))[lane];
  float* d = agg + (size_t)dst * RDIM + lane * 4;
  atomicAdd(d + 0, v.x);
  atomicAdd(d + 1, v.y);
  atomicAdd(d + 2, v.z);
  atomicAdd(d + 3, v.w);
}

// ---------------------------------------------------------------------------
// Kernel 5: o = l2norm(relu((x + agg) @ W + b)) using V_WMMA_F32_16X16X4_F32.
// 625 blocks * 16 rows; 8 waves/block, each wave owns a 16x16 output tile
// (16 rows x 16 of the 128 columns), K=128 via 32 WMMA 16x16x4 steps.
// ---------------------------------------------------------------------------
__global__ __launch_bounds__(256) void psi2_gemm_kernel(const float* __restrict__ x,
                                                        const float* __restrict__ agg,
                                                        const float* __restrict__ W,
                                                        const float* __restrict__ b,
                                                        float* __restrict__ out) {
  __shared__ float hT[16][RDIM + 4];   // +4 pad: kills 16-way bank conflict on column reads
  __shared__ float oT[16][RDIM + 4];

  const int rowBase = blockIdx.x * 16;
  const int t = threadIdx.x;

  // Stage h = x + agg for 16 rows
  for (int e = t; e < 16 * RDIM; e += 256) {
    int r = e >> 7, c = e & (RDIM - 1);
    size_t g = (size_t)(rowBase + r) * RDIM + c;
    hT[r][c] = x[g] + agg[g];
  }
  __syncthreads();

  const int wave = t >> 5;
  const int lane = t & 31;
  const int half = lane >> 4;        // 0: lanes 0-15, 1: lanes 16-31
  const int mrow = lane & 15;
  const int khalf = half << 1;       // 0 or 2
  const int col = wave * 16 + mrow;  // output column this lane covers

  v8f acc = {};
#pragma unroll
  for (int k = 0; k < RDIM; k += 4) {
    v2f a, bf;
    // A (16x4 f32): lanes 0-15 hold K=k,k+1; lanes 16-31 hold K=k+2,k+3
    a.x = hT[mrow][k + khalf];
    a.y = hT[mrow][k + khalf + 1];
    // B (4x16 f32): row-striped across lanes; W[k][n] with n = col
    bf.x = W[(size_t)(k + khalf) * RDIM + col];
    bf.y = W[(size_t)(k + khalf + 1) * RDIM + col];
    acc = __builtin_amdgcn_wmma_f32_16x16x4_f32(false, a, false, bf,
                                                (short)0, acc, false, false);
  }

  // C/D layout: VGPR j -> M = j + 8*half, N = col. Bias + relu into LDS tile.
  float bias = b[col];
#pragma unroll
  for (int j = 0; j < 8; ++j) {
    int m = j + (half << 3);
    float v = acc[j] + bias;
    oT[m][col] = v > 0.0f ? v : 0.0f;
  }
  __syncthreads();

  // Row l2-normalize: each wave handles 2 rows, 4 elems/lane, butterfly reduce.
#pragma unroll
  for (int rr = 0; rr < 2; ++rr) {
    int r = wave * 2 + rr;
    float4 v = *(const float4*)&oT[r][lane * 4];
    float ss = v.x * v.x + v.y * v.y + v.z * v.z + v.w * v.w;
#pragma unroll
    for (int m = 16; m >= 1; m >>= 1) ss += __shfl_xor(ss, m, 32);
    float scale = 1.0f / fmaxf(__fsqrt_rn(ss), 1e-12f);
    float4 o = make_float4(v.x * scale, v.y * scale, v.z * scale, v.w * scale);
    *(float4*)&out[(size_t)(rowBase + r) * RDIM + lane * 4] = o;
  }
}

// ---------------------------------------------------------------------------
// Kernel 6: sim[i,k] = <o_s[i], o_t[knn[i,k]]>; s_hat += sim; S = softmax(s_hat)
// One wave per row i (8 waves / block).
// ---------------------------------------------------------------------------
__global__ __launch_bounds__(256) void sim_softmax_kernel(const float* __restrict__ o_s,
                                                          const float* __restrict__ o_t,
                                                          const int* __restrict__ knn,
                                                          float* __restrict__ s_hat,
                                                          float* __restrict__ S_out) {
  int t = threadIdx.x;
  int wave = t >> 5, lane = t & 31;
  int i = blockIdx.x * 8 + wave;

  float4 os = ((const float4*)(o_s + (size_t)i * RDIM))[lane];

  float sv[KP1];
#pragma unroll
  for (int k = 0; k < KP1; ++k) {
    int j = knn[i * KP1 + k];
    float4 ot = ((const float4*)(o_t + (size_t)j * RDIM))[lane];
    float p = os.x * ot.x + os.y * ot.y + os.z * ot.z + os.w * ot.w;
#pragma unroll
    for (int m = 16; m >= 1; m >>= 1) p += __shfl_xor(p, m, 32);
    float s = s_hat[i * KP1 + k] + p;       // all lanes hold same s after butterfly
    if (lane == 0) s_hat[i * KP1 + k] = s;
    sv[k] = s;
  }

  if (lane == 0) {
    float mx = sv[0];
#pragma unroll
    for (int k = 1; k < KP1; ++k) mx = fmaxf(mx, sv[k]);
    float ex[KP1], sum = 0.0f;
#pragma unroll
    for (int k = 0; k < KP1; ++k) { ex[k] = __expf(sv[k] - mx); sum += ex[k]; }
    float inv = 1.0f / sum;
#pragma unroll
    for (int k = 0; k < KP1; ++k) S_out[i * KP1 + k] = ex[k] * inv;
  }
}

// ---------------------------------------------------------------------------
// Host orchestration
// ---------------------------------------------------------------------------
extern "C" void kernel_launch(void* const* d_in, const int* in_sizes, int n_in,
                              void* d_out, int out_size, void* d_ws, size_t ws_size,
                              hipStream_t stream) {
  const float* S_hat  = (const float*)d_in[0];
  const float* W      = (const float*)d_in[1];
  const float* b      = (const float*)d_in[2];
  const int*   edgesS = (const int*)d_in[3];
  const int*   edgesT = (const int*)d_in[4];
  float* out = (float*)d_out;

  // Workspace carve-up (~27 MB total)
  char* ws = (char*)d_ws;
  size_t off = 0;
  auto carve = [&](size_t bytes) -> void* {
    off = (off + 255) & ~(size_t)255;
    void* p = ws + off;
    off += bytes;
    return p;
  };
  float* s_hat = (float*)carve((size_t)N0V * KP1 * 4);
  int*   knn   = (int*)  carve((size_t)N0V * KP1 * 4);
  float* Sbuf  = (float*)carve((size_t)N0V * KP1 * 4);
  float* r_s   = (float*)carve((size_t)N0V * RDIM * 4);
  float* r_t   = (float*)carve((size_t)N1V * RDIM * 4);
  float* aggb  = (float*)carve((size_t)N0V * RDIM * 4);
  float* o_s   = (float*)carve((size_t)N0V * RDIM * 4);
  float* o_t   = (float*)carve((size_t)N1V * RDIM * 4);
  (void)ws_size; (void)in_sizes; (void)n_in; (void)out_size;

  const size_t featBytes = (size_t)N0V * RDIM * 4;

  // 1) top-(k+1), s_hat = vals*NUM_STEPS, S = softmax(s_hat)
  topk_kernel<<<N0V, 256, 0, stream>>>(S_hat, s_hat, knn, Sbuf);

  for (int step = 0; step < NSTEPS; ++step) {
    // r_s ~ normal(fold_in(key(42), step))
    rng_kernel<<<(N0V * RDIM / 2) / 256, 256, 0, stream>>>(r_s, step);

    // r_t = sp(S)^T @ r_s (scatter over knn)
    hipMemsetAsync(r_t, 0, featBytes, stream);
    scatter_rt_kernel<<<(N0V * KP1 * 32) / 256, 256, 0, stream>>>(Sbuf, knn, r_s, r_t);

    // o_s = l2norm(psi2(r_s, edges_s))
    hipMemsetAsync(aggb, 0, featBytes, stream);
    edge_scatter_kernel<<<(NEDGES * 32) / 256, 256, 0, stream>>>(edgesS, r_s, aggb);
    psi2_gemm_kernel<<<N0V / 16, 256, 0, stream>>>(r_s, aggb, W, b, o_s);

    // o_t = l2norm(psi2(r_t, edges_t))
    hipMemsetAsync(aggb, 0, featBytes, stream);
    edge_scatter_kernel<<<(NEDGES * 32) / 256, 256, 0, stream>>>(edgesT, r_t, aggb);
    psi2_gemm_kernel<<<N1V / 16, 256, 0, stream>>>(r_t, aggb, W, b, o_t);

    // sims + s_hat update + softmax; final step writes straight to d_out
    float* Sdst = (step == NSTEPS - 1) ? out : Sbuf;
    sim_softmax_kernel<<<N0V / 8, 256, 0, stream>>>(o_s, o_t, knn, s_hat, Sdst);
  }
}